// Rec1_43748536877661
// MI455X (gfx1250) — compile-verified
//
#include <hip/hip_runtime.h>
#include <hip/hip_bf16.h>
#include <math.h>

typedef float v2f __attribute__((ext_vector_type(2)));
typedef float v8f __attribute__((ext_vector_type(8)));

#define BB 2
#define SS 2048
#define DD 128
#define NN 2048
#define GG 32           // chunks
#define LL 64           // chunk length (GG*LL == SS)

// ---------------------------------------------------------------------------
// Kernel 1: fused featurizer. One wave per 16x16 output tile of [B*S, N].
// Computes Bf = x*W_B^T + b_B, Cf = x*W_C^T + b_C, dt = softplus(x*W_dt^T+b_dt)
// then a = exp(dt * -exp(A_log)), b = dt*Bf, c = Cf; writes a,b,c to ws.
// ---------------------------------------------------------------------------
__global__ __launch_bounds__(256)
void ssm_featurizer_wmma(const float* __restrict__ x,
                         const float* __restrict__ W_B, const float* __restrict__ b_B,
                         const float* __restrict__ W_C, const float* __restrict__ b_C,
                         const float* __restrict__ W_dt, const float* __restrict__ b_dt,
                         const float* __restrict__ A_log,
                         float* __restrict__ Aarr, float* __restrict__ Barr,
                         float* __restrict__ Carr) {
  const int lane = threadIdx.x & 31;
  const int wid  = (blockIdx.x * blockDim.x + threadIdx.x) >> 5;
  const int NT_N = NN / 16;                 // 128 n-tiles
  const int mtile = wid / NT_N;             // 0..255
  const int ntile = wid - mtile * NT_N;     // 0..127
  const int m0 = mtile * 16;
  const int n0 = ntile * 16;

  const int row = lane & 15;                // M (for A) / N (for B) index
  const int kof = (lane >> 4) * 2;          // k offset 0 or 2 within 4-chunk

  // Preload all A fragments (x tile 16x128): 32 chunks of K=4.
  const float* xrow = x + (size_t)(m0 + row) * DD + kof;
  v2f af[32];
#pragma unroll
  for (int kk = 0; kk < 32; ++kk)
    af[kk] = *reinterpret_cast<const v2f*>(xrow + kk * 4);

  const float* wbrow = W_B  + (size_t)(n0 + row) * DD + kof;
  const float* wcrow = W_C  + (size_t)(n0 + row) * DD + kof;
  const float* wdrow = W_dt + (size_t)(n0 + row) * DD + kof;

  v8f accB = {0.f,0.f,0.f,0.f,0.f,0.f,0.f,0.f};
  v8f accC = accB;
  v8f accD = accB;
#pragma unroll
  for (int kk = 0; kk < 32; ++kk) {
    v2f bf = *reinterpret_cast<const v2f*>(wbrow + kk * 4);
    accB = __builtin_amdgcn_wmma_f32_16x16x4_f32(false, af[kk], false, bf,
                                                 (short)0, accB, false, false);
  }
#pragma unroll
  for (int kk = 0; kk < 32; ++kk) {
    v2f bf = *reinterpret_cast<const v2f*>(wcrow + kk * 4);
    accC = __builtin_amdgcn_wmma_f32_16x16x4_f32(false, af[kk], false, bf,
                                                 (short)0, accC, false, false);
  }
#pragma unroll
  for (int kk = 0; kk < 32; ++kk) {
    v2f bf = *reinterpret_cast<const v2f*>(wdrow + kk * 4);
    accD = __builtin_amdgcn_wmma_f32_16x16x4_f32(false, af[kk], false, bf,
                                                 (short)0, accD, false, false);
  }

  // Elementwise epilogue. C/D layout: M = r + 8*(lane>>4), N = lane&15.
  const int n    = n0 + (lane & 15);
  const int mofs = (lane >> 4) * 8;
  const float bBn  = b_B[n];
  const float bCn  = b_C[n];
  const float bDn  = b_dt[n];
  const float Aneg = -__expf(A_log[n]);

#pragma unroll
  for (int r = 0; r < 8; ++r) {
    const int m = m0 + r + mofs;            // flattened b*S + s
    const size_t idx = (size_t)m * NN + n;
    float z   = accD[r] + bDn;
    float dtv = (z > 20.f) ? z : log1pf(__expf(z));  // softplus
    float a   = __expf(dtv * Aneg);
    Aarr[idx] = a;
    Barr[idx] = dtv * (accB[r] + bBn);
    Carr[idx] = accC[r] + bCn;
  }
}

// ---------------------------------------------------------------------------
// Kernel 2: chunked-scan pass 1. Per (b, g, n): chunk product P and local
// scan terminal value Sv (with h0 = 0).
// ---------------------------------------------------------------------------
__global__ __launch_bounds__(256)
void ssm_scan_pass1(const float* __restrict__ Aarr, const float* __restrict__ Barr,
                    float* __restrict__ Parr, float* __restrict__ Svarr) {
  const int n = blockIdx.x * blockDim.x + threadIdx.x;
  const int g = blockIdx.y;
  const int b = blockIdx.z;
  size_t base = ((size_t)b * SS + (size_t)g * LL) * NN + n;
  float P = 1.f, Sv = 0.f;
#pragma unroll 4
  for (int t = 0; t < LL; ++t) {
    float a  = Aarr[base];
    float bb = Barr[base];
    Sv = a * Sv + bb;
    P *= a;
    base += NN;
  }
  const size_t o = ((size_t)b * GG + g) * NN + n;
  Parr[o]  = P;
  Svarr[o] = Sv;
}

// ---------------------------------------------------------------------------
// Kernel 3: sequential combine across the G=32 chunks (tiny). Per (b, n):
// h_in for each chunk.
// ---------------------------------------------------------------------------
__global__ __launch_bounds__(256)
void ssm_scan_combine(const float* __restrict__ Parr, const float* __restrict__ Svarr,
                      float* __restrict__ Hin) {
  const int n = blockIdx.x * blockDim.x + threadIdx.x;
  const int b = blockIdx.z;
  float h = 0.f;
#pragma unroll
  for (int g = 0; g < GG; ++g) {
    const size_t o = ((size_t)b * GG + g) * NN + n;
    Hin[o] = h;
    h = Parr[o] * h + Svarr[o];
  }
}

// ---------------------------------------------------------------------------
// Kernel 4: pass 3. Re-run each chunk with the correct incoming state,
// producing y = c * h IN PLACE over the Carr buffer.
// ---------------------------------------------------------------------------
__global__ __launch_bounds__(256)
void ssm_scan_pass3(const float* __restrict__ Aarr, const float* __restrict__ Barr,
                    float* __restrict__ Carr /* becomes Y */,
                    const float* __restrict__ Hin) {
  const int n = blockIdx.x * blockDim.x + threadIdx.x;
  const int g = blockIdx.y;
  const int b = blockIdx.z;
  float h = Hin[((size_t)b * GG + g) * NN + n];
  size_t base = ((size_t)b * SS + (size_t)g * LL) * NN + n;
#pragma unroll 4
  for (int t = 0; t < LL; ++t) {
    float a  = Aarr[base];
    float bb = Barr[base];
    float c  = Carr[base];
    h = a * h + bb;
    Carr[base] = c * h;
    base += NN;
  }
}

// ---------------------------------------------------------------------------
// Kernel 5: output projection. out[m,d] = sum_n Y[m,n]*W_out[d,n] + b_out[d].
// One wave per 16x16 tile of [B*S, D]; K = N = 2048 -> 512 fp32 WMMA ops.
// ---------------------------------------------------------------------------
__global__ __launch_bounds__(256)
void ssm_outproj_wmma(const float* __restrict__ Y,
                      const float* __restrict__ W_out, const float* __restrict__ b_out,
                      float* __restrict__ out) {
  const int lane = threadIdx.x & 31;
  const int wid  = (blockIdx.x * blockDim.x + threadIdx.x) >> 5;
  const int NT_D = DD / 16;                 // 8 d-tiles
  const int mtile = wid / NT_D;             // 0..255
  const int dtile = wid - mtile * NT_D;     // 0..7
  const int m0 = mtile * 16;
  const int d0 = dtile * 16;

  const int row = lane & 15;
  const int kof = (lane >> 4) * 2;

  const float* yrow = Y     + (size_t)(m0 + row) * NN + kof;
  const float* wrow = W_out + (size_t)(d0 + row) * NN + kof;

  v8f acc = {0.f,0.f,0.f,0.f,0.f,0.f,0.f,0.f};
#pragma unroll 8
  for (int kk = 0; kk < NN / 4; ++kk) {     // 512 iters
    v2f af = *reinterpret_cast<const v2f*>(yrow + kk * 4);
    v2f bf = *reinterpret_cast<const v2f*>(wrow + kk * 4);
    acc = __builtin_amdgcn_wmma_f32_16x16x4_f32(false, af, false, bf,
                                                (short)0, acc, false, false);
  }

  const int d    = d0 + (lane & 15);
  const int mofs = (lane >> 4) * 8;
  const float bo = b_out[d];
#pragma unroll
  for (int r = 0; r < 8; ++r) {
    const int m = m0 + r + mofs;
    out[(size_t)m * DD + d] = acc[r] + bo;
  }
}

// ---------------------------------------------------------------------------
extern "C" void kernel_launch(void* const* d_in, const int* in_sizes, int n_in,
                              void* d_out, int out_size, void* d_ws, size_t ws_size,
                              hipStream_t stream) {
  const float* x     = (const float*)d_in[0];
  const float* W_B   = (const float*)d_in[1];
  const float* b_B   = (const float*)d_in[2];
  const float* W_C   = (const float*)d_in[3];
  const float* b_C   = (const float*)d_in[4];
  const float* W_dt  = (const float*)d_in[5];
  const float* b_dt  = (const float*)d_in[6];
  const float* A_log = (const float*)d_in[7];
  const float* W_out = (const float*)d_in[8];
  const float* b_out = (const float*)d_in[9];
  float* out = (float*)d_out;

  float* ws = (float*)d_ws;
  const size_t SN = (size_t)BB * SS * NN;   // 8 Mi elements
  const size_t GN = (size_t)BB * GG * NN;   // 128 Ki elements
  float* Aarr  = ws;
  float* Barr  = ws + SN;
  float* Carr  = ws + 2 * SN;               // becomes Y in-place
  float* Parr  = ws + 3 * SN;
  float* Svarr = Parr + GN;
  float* Hin   = Svarr + GN;

  // K1: (B*S/16)*(N/16) = 32768 waves, 8 waves/block -> 4096 blocks
  ssm_featurizer_wmma<<<4096, 256, 0, stream>>>(x, W_B, b_B, W_C, b_C, W_dt, b_dt,
                                                A_log, Aarr, Barr, Carr);
  // K2: pass 1 over chunks
  ssm_scan_pass1<<<dim3(NN / 256, GG, BB), 256, 0, stream>>>(Aarr, Barr, Parr, Svarr);
  // K3: combine (B*N threads)
  ssm_scan_combine<<<dim3(NN / 256, 1, BB), 256, 0, stream>>>(Parr, Svarr, Hin);
  // K4: pass 3, y written in place over Carr
  ssm_scan_pass3<<<dim3(NN / 256, GG, BB), 256, 0, stream>>>(Aarr, Barr, Carr, Hin);
  // K5: (B*S/16)*(D/16) = 2048 waves -> 256 blocks
  ssm_outproj_wmma<<<256, 256, 0, stream>>>(Carr, W_out, b_out, out);
}